// AttentionLayer_35029753266764
// MI455X (gfx1250) — compile-verified
//
#include <hip/hip_runtime.h>
#include <hip/hip_bf16.h>
#include <math.h>

// Problem constants
#define BB    64
#define NN    2048
#define DD    256       // DIM
#define HH    4         // HEADS
#define DHD   64        // DH
#define SCALE 0.125f    // 64^-0.5

typedef __attribute__((ext_vector_type(2))) float v2f;
typedef __attribute__((ext_vector_type(8))) float v8f;

// ---------------------------------------------------------------------------
// Precompute 1: Wvo = Wv @ Wo  (256x1024 @ 1024x256), bvo = bv@Wo + bo
// grid 256 (d), block 256 (e)
__global__ void k_wvo(const float* __restrict__ Wv, const float* __restrict__ Wo,
                      const float* __restrict__ bv, const float* __restrict__ bo,
                      float* __restrict__ Wvo, float* __restrict__ bvo) {
    int d = blockIdx.x, e = threadIdx.x;
    float acc = 0.f;
    for (int j = 0; j < HH * DD; ++j)
        acc += Wv[(size_t)d * (HH * DD) + j] * Wo[(size_t)j * DD + e];
    Wvo[(size_t)d * DD + e] = acc;
    if (d == 0) {
        float b = bo[e];
        for (int j = 0; j < HH * DD; ++j)
            b += bv[j] * Wo[(size_t)j * DD + e];
        bvo[e] = b;
    }
}

// Precompute 2: Q[b, i] = x[b,0,:] @ Wq    grid 64 (b), block 256 (i)
__global__ void k_q(const float* __restrict__ x, const float* __restrict__ Wq,
                    float* __restrict__ Q) {
    int b = blockIdx.x, i = threadIdx.x;
    const float* x0 = x + (size_t)b * NN * DD;
    float acc = 0.f;
    for (int d = 0; d < DD; ++d)
        acc += x0[d] * Wq[(size_t)d * DD + i];
    Q[(size_t)b * DD + i] = acc;
}

// Precompute 3: QK[b,h,d] = sum_e Wk[d, h*64+e] * Q[b, h*64+e]
// grid 256 (b*4+h), block 256 (d)
__global__ void k_qk(const float* __restrict__ Wk, const float* __restrict__ Q,
                     float* __restrict__ QK) {
    int b = blockIdx.x >> 2, h = blockIdx.x & 3;
    int d = threadIdx.x;
    float acc = 0.f;
    for (int e = 0; e < DHD; ++e)
        acc += Wk[(size_t)d * DD + h * DHD + e] * Q[(size_t)b * DD + h * DHD + e];
    QK[((size_t)b * HH + h) * DD + d] = acc;
}

// ---------------------------------------------------------------------------
// dots[b,h,n] = SCALE * x[b,n,:] . QK[b,h,:]
// grid (32 chunks, 64 b), block 256 = 8 waves, each wave does 8 tokens
__global__ void k_dots(const float* __restrict__ x, const float* __restrict__ QK,
                       float* __restrict__ dots) {
    int b = blockIdx.y;
    __shared__ float qks[HH * DD];
    for (int i = threadIdx.x; i < HH * DD; i += 256)
        qks[i] = QK[(size_t)b * HH * DD + i];
    __syncthreads();
    int wave = threadIdx.x >> 5, lane = threadIdx.x & 31;
    const float* X = x + (size_t)b * NN * DD;
    for (int i = 0; i < 8; ++i) {
        int n = blockIdx.x * 64 + i * 8 + wave;
        const float* xp = X + (size_t)n * DD + lane * 8;
        float4 x0 = *(const float4*)xp;
        float4 x1 = *(const float4*)(xp + 4);
        float p[HH];
#pragma unroll
        for (int h = 0; h < HH; ++h) {
            const float* qp = qks + h * DD + lane * 8;
            p[h] = x0.x * qp[0] + x0.y * qp[1] + x0.z * qp[2] + x0.w * qp[3] +
                   x1.x * qp[4] + x1.y * qp[5] + x1.z * qp[6] + x1.w * qp[7];
        }
#pragma unroll
        for (int off = 16; off > 0; off >>= 1)
#pragma unroll
            for (int h = 0; h < HH; ++h)
                p[h] += __shfl_xor(p[h], off, 32);
        if (lane == 0)
#pragma unroll
            for (int h = 0; h < HH; ++h)
                dots[((size_t)b * HH + h) * NN + n] = p[h] * SCALE;
    }
}

// ---------------------------------------------------------------------------
// mean-threshold mask + softmax, in place (dots -> attn). grid 256 (b*h), block 256
__global__ void k_softmax(float* __restrict__ dots) {
    float* d = dots + (size_t)blockIdx.x * NN;
    __shared__ float red[256];
    int t = threadIdx.x;
    float v[8];
    float lsum = 0.f;
#pragma unroll
    for (int i = 0; i < 8; ++i) { v[i] = d[t + i * 256]; lsum += v[i]; }
    red[t] = lsum; __syncthreads();
    for (int s = 128; s > 0; s >>= 1) { if (t < s) red[t] += red[t + s]; __syncthreads(); }
    float mean = red[0] * (1.0f / NN); __syncthreads();

    bool keep[8];
    float lmax = -INFINITY;
#pragma unroll
    for (int i = 0; i < 8; ++i) {
        int n = t + i * 256;
        keep[i] = (v[i] >= mean) || (n == 0);
        if (keep[i]) lmax = fmaxf(lmax, v[i]);
    }
    red[t] = lmax; __syncthreads();
    for (int s = 128; s > 0; s >>= 1) { if (t < s) red[t] = fmaxf(red[t], red[t + s]); __syncthreads(); }
    float mx = red[0]; __syncthreads();

    float e[8];
    float lexp = 0.f;
#pragma unroll
    for (int i = 0; i < 8; ++i) { e[i] = keep[i] ? expf(v[i] - mx) : 0.f; lexp += e[i]; }
    red[t] = lexp; __syncthreads();
    for (int s = 128; s > 0; s >>= 1) { if (t < s) red[t] += red[t + s]; __syncthreads(); }
    float inv = 1.f / red[0];
#pragma unroll
    for (int i = 0; i < 8; ++i) d[t + i * 256] = e[i] * inv;
}

// ---------------------------------------------------------------------------
__global__ void k_zero(float* __restrict__ p, int n) {
    int i = blockIdx.x * blockDim.x + threadIdx.x;
    if (i < n) p[i] = 0.f;
}

// s[b,h,d] = sum_n attn[b,h,n] * x[b,n,d]   grid (8 chunks, 64 b), block 256 (d)
__global__ void k_accum_s(const float* __restrict__ x, const float* __restrict__ attn,
                          float* __restrict__ s) {
    int b = blockIdx.y, chunk = blockIdx.x;
    int t = threadIdx.x;
    __shared__ float aw[HH * 256];
#pragma unroll
    for (int h = 0; h < HH; ++h)
        aw[h * 256 + t] = attn[((size_t)b * HH + h) * NN + chunk * 256 + t];
    __syncthreads();
    const float* X = x + (size_t)b * NN * DD + (size_t)chunk * 256 * DD;
    float a0 = 0.f, a1 = 0.f, a2 = 0.f, a3 = 0.f;
    for (int n = 0; n < 256; ++n) {
        float xv = X[(size_t)n * DD + t];
        a0 += aw[n] * xv;
        a1 += aw[256 + n] * xv;
        a2 += aw[512 + n] * xv;
        a3 += aw[768 + n] * xv;
    }
    atomicAdd(&s[((size_t)b * HH + 0) * DD + t], a0);
    atomicAdd(&s[((size_t)b * HH + 1) * DD + t], a1);
    atomicAdd(&s[((size_t)b * HH + 2) * DD + t], a2);
    atomicAdd(&s[((size_t)b * HH + 3) * DD + t], a3);
}

// t[b, h*256+c] = s[b,h,:] @ Wv[:, h*256+c] + bv[h*256+c]   grid 256 (b*4+h), block 256 (c)
__global__ void k_token(const float* __restrict__ s, const float* __restrict__ Wv,
                        const float* __restrict__ bv, float* __restrict__ tk) {
    int b = blockIdx.x >> 2, h = blockIdx.x & 3;
    int c = threadIdx.x;
    int j = h * DD + c;
    float acc = bv[j];
    for (int d = 0; d < DD; ++d)
        acc += s[((size_t)b * HH + h) * DD + d] * Wv[(size_t)d * (HH * DD) + j];
    tk[(size_t)b * (HH * DD) + j] = acc;
}

// out[b,0,e] = tk[b,:] @ Wo + bo   grid 64 (b), block 256 (e)
__global__ void k_out0(const float* __restrict__ tk, const float* __restrict__ Wo,
                       const float* __restrict__ bo, float* __restrict__ out) {
    int b = blockIdx.x, e = threadIdx.x;
    float acc = bo[e];
    for (int j = 0; j < HH * DD; ++j)
        acc += tk[(size_t)b * (HH * DD) + j] * Wo[(size_t)j * DD + e];
    out[(size_t)b * NN * DD + e] = acc;
}

// ---------------------------------------------------------------------------
// Main GEMM: out[b,n,:] = x[b,n,:] @ Wvo + bvo    (exact f32 via V_WMMA_F32_16X16X4_F32)
// grid (32 row-blocks of 64, 64 b), block 256 = 8 waves.
// Each wave: 32 rows x 64 cols = 2 row tiles x 4 col tiles, B fragments reused
// in registers across both row tiles (1.25 load-issues per WMMA).
__global__ void k_gemm_out(const float* __restrict__ x, const float* __restrict__ Wvo,
                           const float* __restrict__ bvo, float* __restrict__ out) {
    int b = blockIdx.y;
    int wave = threadIdx.x >> 5;
    int lane = threadIdx.x & 31;
    int lhalf = lane >> 4;          // 0: lanes 0-15 (K=k,k+1)  1: lanes 16-31 (K=k+2,k+3)
    int lmod  = lane & 15;
    int r0 = blockIdx.x * 64 + (wave >> 2) * 32;   // 32-row strip (2 tiles)
    int c0 = (wave & 3) * 64;                       // 64-col strip (4 WMMA tiles)
    const float* X = x + (size_t)b * NN * DD;
    float* O = out + (size_t)b * NN * DD;

    v8f acc[2][4] = {};
    for (int k = 0; k < DD; k += 4) {
        // A fragments for both row tiles: 16x4 f32 each
        const float* apl = X + (size_t)(r0 + lmod) * DD + k + 2 * lhalf;
        const float* aph = apl + (size_t)16 * DD;
        v2f al = { apl[0], apl[1] };
        v2f ah = { aph[0], aph[1] };
        // B fragments: 4x16 f32 (vgpr0 = rows k,k+2 ; vgpr1 = rows k+1,k+3)
        int krow = k + 2 * lhalf;
        const float* bp0 = Wvo + (size_t)krow * DD + c0 + lmod;
        const float* bp1 = bp0 + DD;
        v2f bf[4];
#pragma unroll
        for (int s = 0; s < 4; ++s) {
            v2f tmp = { bp0[16 * s], bp1[16 * s] };
            bf[s] = tmp;
        }
#pragma unroll
        for (int s = 0; s < 4; ++s) {
            acc[0][s] = __builtin_amdgcn_wmma_f32_16x16x4_f32(false, al, false, bf[s],
                                                              (short)0, acc[0][s], false, false);
            acc[1][s] = __builtin_amdgcn_wmma_f32_16x16x4_f32(false, ah, false, bf[s],
                                                              (short)0, acc[1][s], false, false);
        }
    }
    // C/D layout: vgpr j -> row tile_r0 + j + 8*lhalf, col = tile_c0 + lmod
#pragma unroll
    for (int g = 0; g < 2; ++g) {
#pragma unroll
        for (int j = 0; j < 8; ++j) {
            int r = r0 + g * 16 + j + 8 * lhalf;
            size_t rowoff = (size_t)r * DD;
            int cb = c0 + lmod;
#pragma unroll
            for (int s = 0; s < 4; ++s)
                O[rowoff + cb + 16 * s] = acc[g][s][j] + bvo[cb + 16 * s];
        }
    }
}

// ---------------------------------------------------------------------------
extern "C" void kernel_launch(void* const* d_in, const int* in_sizes, int n_in,
                              void* d_out, int out_size, void* d_ws, size_t ws_size,
                              hipStream_t stream) {
    const float* x  = (const float*)d_in[0];
    const float* Wq = (const float*)d_in[1];
    const float* Wk = (const float*)d_in[2];
    const float* Wv = (const float*)d_in[3];
    const float* bv = (const float*)d_in[4];
    const float* Wo = (const float*)d_in[5];
    const float* bo = (const float*)d_in[6];
    float* out = (float*)d_out;

    // workspace layout (bytes, 256-aligned)
    char* ws = (char*)d_ws;
    float* Wvo  = (float*)(ws + 0);            // 256*256*4      = 262144
    float* bvo  = (float*)(ws + 262144);       // 256*4          = 1024
    float* Q    = (float*)(ws + 263424);       // 64*256*4       = 65536
    float* QK   = (float*)(ws + 328960);       // 64*4*256*4     = 262144
    float* dots = (float*)(ws + 591104);       // 64*4*2048*4    = 2097152  (becomes attn)
    float* S    = (float*)(ws + 2688256);      // 64*4*256*4     = 262144
    float* TK   = (float*)(ws + 2950400);      // 64*1024*4      = 262144
    (void)ws_size; (void)in_sizes; (void)n_in; (void)out_size;

    // Precompute fused weights + query projections
    k_wvo<<<dim3(DD), dim3(DD), 0, stream>>>(Wv, Wo, bv, bo, Wvo, bvo);
    k_q  <<<dim3(BB), dim3(DD), 0, stream>>>(x, Wq, Q);
    k_qk <<<dim3(BB * HH), dim3(DD), 0, stream>>>(Wk, Q, QK);

    // Attention scores, mask, softmax
    k_dots   <<<dim3(NN / 64, BB), dim3(256), 0, stream>>>(x, QK, dots);
    k_softmax<<<dim3(BB * HH), dim3(256), 0, stream>>>(dots);

    // s = attn . x  (zero-init then atomic partial sums)
    k_zero   <<<dim3((BB * HH * DD + 255) / 256), dim3(256), 0, stream>>>(S, BB * HH * DD);
    k_accum_s<<<dim3(NN / 256, BB), dim3(256), 0, stream>>>(x, dots, S);

    // token projection (two tiny GEMMs)
    k_token<<<dim3(BB * HH), dim3(DD), 0, stream>>>(S, Wv, bv, TK);

    // bulk output GEMM (writes all rows incl. n=0), then overwrite row 0
    k_gemm_out<<<dim3(NN / 64, BB), dim3(256), 0, stream>>>(x, Wvo, bvo, out);
    k_out0    <<<dim3(BB), dim3(DD), 0, stream>>>(TK, Wo, bo, out);
}